// Group_77412490543802
// MI455X (gfx1250) — compile-verified
//
#include <hip/hip_runtime.h>
#include <hip/hip_bf16.h>

// Problem constants (from reference):
#define BATCH 16
#define NPTS  16384
#define NGRP  1024
#define GSIZE 32
#define BIGF  1e10f

// KNN chunking
#define CH        64          // ref points per chunk
#define NCH       (NPTS / CH) // 256 chunks
#define ROWSTRIDE 68          // d2 row stride (dwords); halves at +0 / +34 -> conflict-free
#define HALF1OFF  34

typedef __attribute__((ext_vector_type(2))) float v2f;
typedef __attribute__((ext_vector_type(8))) float v8f;

#define AS1 __attribute__((address_space(1)))
#define AS3 __attribute__((address_space(3)))

#if defined(__AMDGCN__) && __has_builtin(__builtin_amdgcn_global_load_async_to_lds_b32)
#define USE_ASYNC 1
#else
#define USE_ASYNC 0
#endif

// ---------------------------------------------------------------------------
// Kernel 1: Farthest Point Sampling.
// One 1024-thread workgroup per batch; each thread owns 16 points entirely in
// registers (coords + running min-distance). Per step: register distance
// update -> wave32 shfl_xor butterfly argmax -> 32-partial LDS tree ->
// owner thread publishes the next centroid coords. Centers -> d_out.
// ---------------------------------------------------------------------------
__global__ __launch_bounds__(1024) void fps_kernel(const float* __restrict__ xyz,
                                                   float* __restrict__ centers)
{
    __shared__ float red_v[32];
    __shared__ int   red_i[32];
    __shared__ int   s_far;
    __shared__ float s_c[3];

    const int b = blockIdx.x;
    const int t = threadIdx.x;
    const float* base = xyz + (size_t)b * NPTS * 6;

    float px[16], py[16], pz[16], pd[16];
#pragma unroll
    for (int k = 0; k < 16; ++k) {
        const int i = t + k * 1024;
        const float* p = base + (size_t)i * 6;
        px[k] = p[0]; py[k] = p[1]; pz[k] = p[2];
        pd[k] = BIGF;
    }
    if (t == 0) {
        s_far = 0;
        s_c[0] = px[0]; s_c[1] = py[0]; s_c[2] = pz[0]; // point 0 owned by t==0,k==0
    }
    __syncthreads();

    for (int g = 0; g < NGRP; ++g) {
        const float cx = s_c[0], cy = s_c[1], cz = s_c[2];
        if (t == 0) {
            float* c = centers + ((size_t)b * NGRP + g) * 3;
            c[0] = cx; c[1] = cy; c[2] = cz;
        }

        float bestv = -1.0f;
        int   besti = 0;
#pragma unroll
        for (int k = 0; k < 16; ++k) {
            const int i = t + k * 1024;
            const float dx = px[k] - cx, dy = py[k] - cy, dz = pz[k] - cz;
            const float d  = dx * dx + dy * dy + dz * dz;
            const float nd = (d < pd[k]) ? d : pd[k];
            pd[k] = nd;
            if (nd > bestv) { bestv = nd; besti = i; } // strict > keeps first idx
        }
#pragma unroll
        for (int off = 16; off >= 1; off >>= 1) {
            const float ov = __shfl_xor(bestv, off, 32);
            const int   oi = __shfl_xor(besti, off, 32);
            if (ov > bestv || (ov == bestv && oi < besti)) { bestv = ov; besti = oi; }
        }
        if ((t & 31) == 0) { red_v[t >> 5] = bestv; red_i[t >> 5] = besti; }
        __syncthreads();
        if (t < 32) {
            bestv = red_v[t]; besti = red_i[t];
#pragma unroll
            for (int off = 16; off >= 1; off >>= 1) {
                const float ov = __shfl_xor(bestv, off, 32);
                const int   oi = __shfl_xor(besti, off, 32);
                if (ov > bestv || (ov == bestv && oi < besti)) { bestv = ov; besti = oi; }
            }
            if (t == 0) s_far = besti;
        }
        __syncthreads();
        const int far = s_far;
        if ((far & 1023) == t) {
            const int kk = far >> 10;
#pragma unroll
            for (int k = 0; k < 16; ++k) {
                if (k == kk) { s_c[0] = px[k]; s_c[1] = py[k]; s_c[2] = pz[k]; }
            }
        }
        __syncthreads();
    }
}

// ---------------------------------------------------------------------------
// Kernel 2: KNN (top-32 by squared distance) + gather + center-subtract.
// Block = 256 threads = 8 waves; each wave owns 16 queries. Ref points stream
// through a double-buffered LDS chunk (async copies issued by wave 0, tracked
// with ASYNCcnt, one chunk ahead). Per chunk the wave computes the 16x64 dot
// tile with four V_WMMA_F32_16X16X4_F32; all 32 lanes scan (lane l: row l&15,
// column half l>>4) keeping a sorted 32-entry (d2,idx) register list; at the
// end lane pairs merge their lists with a shfl_xor(16) bitonic merge.
// Ordering is ascending (d2, idx) lexicographic == lax.top_k(-d2, 32).
// ---------------------------------------------------------------------------
__global__ __launch_bounds__(256) void knn_group_kernel(const float* __restrict__ xyz,
                                                        const float* __restrict__ centers,
                                                        float* __restrict__ out_xyz,
                                                        float* __restrict__ out_nrm)
{
    __shared__ float s_p[2][3][CH];               // [buf][coord][point]
    __shared__ float s_qq[128];
    __shared__ float s_d2[8 * 16 * ROWSTRIDE];

    const int b     = blockIdx.x >> 3;            // 8 blocks per batch
    const int qbase = (blockIdx.x & 7) * 128;     // 128 queries per block
    const int t     = threadIdx.x;
    const int wave  = t >> 5;
    const int lane  = t & 31;
    const int mhalf = lane >> 4;                  // WMMA D rows 0..7 | 8..15
    const int nn    = lane & 15;

    const float* pbase = xyz + (size_t)b * NPTS * 6;

    // A-fragment (ISA 7.12.2, 16x4 f32): lanes 0-15 carry K=0,1; lanes 16-31 K=2,3
    const int gq = qbase + wave * 16 + nn;
    const float* cp = centers + ((size_t)b * NGRP + gq) * 3;
    const float qx = cp[0], qy = cp[1], qz = cp[2];
    if (lane < 16) s_qq[wave * 16 + lane] = qx * qx + qy * qy + qz * qz;

    v2f afrag;
    afrag.x = (lane < 16) ? qx : qz;   // K=0 | K=2
    afrag.y = (lane < 16) ? qy : 0.0f; // K=1 | K=3

    // sorted register list, ascending (d2, idx)
    float ld[GSIZE];
    int   li[GSIZE];
#pragma unroll
    for (int j = 0; j < GSIZE; ++j) { ld[j] = 3.4e38f; li[j] = 0x7FFFFFFF; }

    auto insert = [&](float d, int id) {
        const bool ins = (d < ld[GSIZE - 1]) ||
                         (d == ld[GSIZE - 1] && id < li[GSIZE - 1]);
        if (ins) {
            ld[GSIZE - 1] = d; li[GSIZE - 1] = id;
#pragma unroll
            for (int j = GSIZE - 1; j >= 1; --j) {
                const bool sw = (ld[j] < ld[j - 1]) ||
                                (ld[j] == ld[j - 1] && li[j] < li[j - 1]);
                const float td = ld[j - 1]; const int ti = li[j - 1];
                if (sw) { ld[j - 1] = ld[j]; li[j - 1] = li[j]; ld[j] = td; li[j] = ti; }
            }
        }
    };

#if USE_ASYNC
    auto prefetch = [&](int c, int buf) {  // executed by wave 0 only
        const float* src = pbase + (size_t)c * CH * 6;
#pragma unroll
        for (int grp = 0; grp < 2; ++grp) {
            const int p = grp * 32 + lane;
            const float* g = src + p * 6;
#pragma unroll
            for (int coord = 0; coord < 3; ++coord) {
                __builtin_amdgcn_global_load_async_to_lds_b32(
                    (AS1 int*)(g + coord),
                    (AS3 int*)&s_p[buf][coord][p], 0, 0);
            }
        }
    };
    if (wave == 0) prefetch(0, 0);
#endif

    float* myd2 = s_d2 + wave * (16 * ROWSTRIDE);
    const float* myrow = myd2 + (lane & 15) * ROWSTRIDE + mhalf * HALF1OFF;

    for (int c = 0; c < NCH; ++c) {
        const int buf = c & 1;
        __syncthreads();                          // previous chunk fully consumed
#if USE_ASYNC
        if (wave == 0) {
            const int pc = (c + 1 < NCH) ? c + 1 : c;   // keep 6 ops in flight
            prefetch(pc, (c + 1) & 1);
#if __has_builtin(__builtin_amdgcn_s_wait_asynccnt)
            __builtin_amdgcn_s_wait_asynccnt(6);        // chunk c landed (in-order)
#else
            asm volatile("s_wait_asynccnt 6" ::: "memory");
#endif
        }
#else
        if (t < CH) {
            const int i = c * CH + t;
            const float* p = pbase + (size_t)i * 6;
            s_p[buf][0][t] = p[0]; s_p[buf][1][t] = p[1]; s_p[buf][2][t] = p[2];
        }
#endif
        __syncthreads();                          // chunk c visible to all waves

        // 16x64 dot tile via 4 WMMAs (uniform control flow: EXEC all ones)
#pragma unroll
        for (int n0 = 0; n0 < CH; n0 += 16) {
            const float rx = s_p[buf][0][n0 + nn];
            const float ry = s_p[buf][1][n0 + nn];
            const float rz = s_p[buf][2][n0 + nn];
            const float rr = rx * rx + ry * ry + rz * rz;
            v2f bfrag;
            bfrag.x = (lane < 16) ? rx : rz;      // K=0 | K=2
            bfrag.y = (lane < 16) ? ry : 0.0f;    // K=1 | K=3
            v8f acc = {};
            acc = __builtin_amdgcn_wmma_f32_16x16x4_f32(
                false, afrag, false, bfrag, (short)0, acc, false, false);
            const int coff = n0 + ((n0 >= 32) ? (HALF1OFF - 32) : 0) + nn;
#pragma unroll
            for (int r = 0; r < 8; ++r) {
                const int m = r + 8 * mhalf;
                const float d2 = s_qq[wave * 16 + m] + rr - 2.0f * acc[r];
                myd2[m * ROWSTRIDE + coff] = d2;
            }
        }
        __syncthreads();

        // selection: all 32 lanes scan; lane l -> row l&15, columns half l>>4
        const int idb = c * CH + mhalf * 32;
        const v2f* row2 = (const v2f*)myrow;      // 8B-aligned b64 LDS loads
        for (int j2 = 0; j2 < 32; j2 += 2) {
            const v2f v = row2[j2 >> 1];
            insert(v.x, idb + j2);
            insert(v.y, idb + j2 + 1);
        }
    }

    // merge lane-pair lists: one cross-lane min-exchange vs reversed partner
    // list (pair j <-> 31-j so both reads precede both writes), leaving the 32
    // global minima as a bitonic sequence in lanes 0-15 ...
#pragma unroll
    for (int j = 0; j < 16; ++j) {
        const float pda = __shfl_xor(ld[31 - j], 16, 32);
        const int   pia = __shfl_xor(li[31 - j], 16, 32);
        const float pdb = __shfl_xor(ld[j], 16, 32);
        const int   pib = __shfl_xor(li[j], 16, 32);
        if (pda < ld[j] || (pda == ld[j] && pia < li[j])) { ld[j] = pda; li[j] = pia; }
        if (pdb < ld[31 - j] || (pdb == ld[31 - j] && pib < li[31 - j])) {
            ld[31 - j] = pdb; li[31 - j] = pib;
        }
    }
    // ... then a 5-stage bitonic clean-up network sorts it ascending.
#pragma unroll
    for (int k = 16; k >= 1; k >>= 1) {
#pragma unroll
        for (int j = 0; j < 32; ++j) {
            if ((j & k) == 0) {
                const int j2 = j + k;
                const bool sw = (ld[j2] < ld[j]) || (ld[j2] == ld[j] && li[j2] < li[j]);
                const float td = ld[j]; const int ti = li[j];
                if (sw) { ld[j] = ld[j2]; li[j] = li[j2]; ld[j2] = td; li[j2] = ti; }
            }
        }
    }

    // gather neighbors + subtract center; normals verbatim
    if (lane < 16) {
        const size_t obase = ((size_t)(b * NGRP + gq)) * GSIZE;
#pragma unroll
        for (int j = 0; j < GSIZE; ++j) {
            const int id = li[j];
            const float* p = pbase + (size_t)id * 6;
            float* o0 = out_xyz + (obase + j) * 3;
            float* o1 = out_nrm + (obase + j) * 3;
            o0[0] = p[0] - qx; o0[1] = p[1] - qy; o0[2] = p[2] - qz;
            o1[0] = p[3];      o1[1] = p[4];      o1[2] = p[5];
        }
    }
}

extern "C" void kernel_launch(void* const* d_in, const int* in_sizes, int n_in,
                              void* d_out, int out_size, void* d_ws, size_t ws_size,
                              hipStream_t stream) {
    (void)in_sizes; (void)n_in; (void)out_size; (void)d_ws; (void)ws_size;
    const float* xyz = (const float*)d_in[0];
    float* out = (float*)d_out;

    const size_t NB3 = (size_t)BATCH * NGRP * GSIZE * 3; // one neighborhood tensor
    float* out_xyz  = out;              // neighborhood_no_normal  [16,1024,32,3]
    float* out_nrm  = out + NB3;        // neighborhood_only_normal[16,1024,32,3]
    float* centers  = out + 2 * NB3;    // center                  [16,1024,3]

    fps_kernel<<<BATCH, 1024, 0, stream>>>(xyz, centers);
    knn_group_kernel<<<BATCH * 8, 256, 0, stream>>>(xyz, centers, out_xyz, out_nrm);
}